// Edge_node_50869592655541
// MI455X (gfx1250) — compile-verified
//
#include <hip/hip_runtime.h>

typedef float v2f __attribute__((ext_vector_type(2)));
typedef float v8f __attribute__((ext_vector_type(8)));

#define ASTRIDE 452   // 448 + 4 pad -> lane banks differ by 4 (conflict-free)
#define BSTRIDE 132   // 128 + 4 pad

// ---------------------------------------------------------------- utilities

__global__ void zerof(float* __restrict__ p, long n) {
  long i = (long)blockIdx.x * blockDim.x + threadIdx.x;
  if (i < n) p[i] = 0.f;
}

// scatter pass 1: deg, S2 (segsum of edge_rep), O (segsum of partner edge rows),
//                 Q (segsum of partner node reps)
__global__ void scatter1(const float* __restrict__ edge_rep,
                         const float* __restrict__ node_rep,
                         const int* __restrict__ eidx, int E, long E2,
                         float* __restrict__ deg, float* __restrict__ S2,
                         float* __restrict__ O, float* __restrict__ Q) {
  long idx = (long)blockIdx.x * blockDim.x + threadIdx.x;
  if (idx >= E2 * 64) return;
  long r = idx >> 6;
  int d = (int)(idx & 63);
  int e = (int)(r >> 1), i = (int)(r & 1);
  int u  = eidx[(size_t)i * E + e];
  int up = eidx[(size_t)(i ^ 1) * E + e];
  long p = r ^ 1;
  atomicAdd(&S2[(size_t)u * 64 + d], edge_rep[r * 64 + d]);
  atomicAdd(&O [(size_t)u * 64 + d], edge_rep[p * 64 + d]);
  atomicAdd(&Q [(size_t)u * 64 + d], node_rep[(size_t)up * 64 + d]);
  if (d == 0) atomicAdd(&deg[u], 1.0f);
}

// scatter pass 2 (needs S2): P[v] = sum over rows at v of S2[partner-node]
__global__ void scatter2(const int* __restrict__ eidx, int E, long E2,
                         const float* __restrict__ S2, float* __restrict__ P) {
  long idx = (long)blockIdx.x * blockDim.x + threadIdx.x;
  if (idx >= E2 * 64) return;
  long r = idx >> 6;
  int d = (int)(idx & 63);
  int e = (int)(r >> 1), i = (int)(r & 1);
  int u  = eidx[(size_t)i * E + e];
  int up = eidx[(size_t)(i ^ 1) * E + e];
  atomicAdd(&P[(size_t)u * 64 + d], S2[(size_t)up * 64 + d]);
}

// ------------------------------------------------- WMMA core (fp32 16x16x4)
// One workgroup = 8 waves; wave w computes rows [rowBase+16w, +16) x Bn cols.
// A-tile is in LDS (ldt-float row stride); B (weights) streamed from global.
template <int NT>
__device__ __forceinline__ void wmma_tile_compute(const float* __restrict__ tile,
                                                  int ldt, int Kdim,
                                                  const float* __restrict__ B,
                                                  float* __restrict__ Y,
                                                  long rowBase, long M) {
  const int Bn = NT * 16;
  const int lane = threadIdx.x & 31;
  const int w = threadIdx.x >> 5;
  const int half = lane >> 4;  // lanes 16-31 hold K+2,K+3 of A / B fragments
  const int l16 = lane & 15;

  v8f acc[NT];
#pragma unroll
  for (int nt = 0; nt < NT; ++nt) acc[nt] = v8f{0.f, 0.f, 0.f, 0.f, 0.f, 0.f, 0.f, 0.f};

  const float* arow = tile + (size_t)(w * 16 + l16) * ldt + half * 2;
  for (int kk = 0; kk < Kdim; kk += 4) {
    v2f a;
    a.x = arow[kk];
    a.y = arow[kk + 1];
    const float* bp = B + (size_t)(kk + half * 2) * Bn + l16;
#pragma unroll
    for (int nt = 0; nt < NT; ++nt) {
      v2f b;
      b.x = bp[nt * 16];
      b.y = bp[nt * 16 + Bn];
      acc[nt] = __builtin_amdgcn_wmma_f32_16x16x4_f32(
          /*neg_a=*/false, a, /*neg_b=*/false, b,
          /*c_mod=*/(short)0, acc[nt], /*reuse_a=*/false, /*reuse_b=*/false);
    }
  }

#pragma unroll
  for (int nt = 0; nt < NT; ++nt) {
#pragma unroll
    for (int j = 0; j < 8; ++j) {
      long row = rowBase + w * 16 + half * 8 + j;
      if (row < M) Y[row * Bn + nt * 16 + l16] = acc[nt][j];
    }
  }
}

// ---------------------------------------- GEMM1 (edge): build fused A-tile
__global__ void edge_gemm1(const float* __restrict__ edge_rep,
                           const float* __restrict__ node_rep,
                           const int* __restrict__ eidx, int E,
                           const float* __restrict__ deg,
                           const float* __restrict__ S2,
                           const float* __restrict__ O,
                           const float* __restrict__ W1,
                           float* __restrict__ He, long M) {
  extern __shared__ float tile[];  // 128 x ASTRIDE floats
  long rowBase = (long)blockIdx.x * 128;
  for (int idx = threadIdx.x; idx < 128 * 64; idx += 256) {
    int q = idx >> 6, d = idx & 63;
    long r = rowBase + q;
    float* Arow = &tile[(size_t)q * ASTRIDE];
    if (r < M) {
      int e = (int)(r >> 1), i = (int)(r & 1);
      int u  = eidx[(size_t)i * E + e];
      int up = eidx[(size_t)(i ^ 1) * E + e];
      float er  = edge_rep[r * 64 + d];
      float dg  = deg[u];
      float s2u = S2[(size_t)u * 64 + d];
      float ou  = O[(size_t)u * 64 + d];
      float s2p = S2[(size_t)up * 64 + d];
      float nu  = node_rep[(size_t)u * 64 + d];
      float np  = node_rep[(size_t)up * 64 + d];
      float t0 = dg * s2u, t1 = t0 + ou, n0 = dg * nu;
      Arow[d]        = er;                 // edge_rep
      Arow[64 + d]   = t0;                 // e2e block 0  (T[u] lo)
      Arow[128 + d]  = t1;                 // e2e block 1  (T[u] hi)
      Arow[192 + d]  = t0 + s2p;           // e2e block 2
      Arow[256 + d]  = t1 + s2p + er;      // e2e block 3
      Arow[320 + d]  = n0;                 // n2e id part
      Arow[384 + d]  = n0 + np;            // n2e bc part
    } else {
#pragma unroll
      for (int b = 0; b < 7; ++b) Arow[b * 64 + d] = 0.f;
    }
  }
  __syncthreads();
  wmma_tile_compute<8>(tile, ASTRIDE, 448, W1, He, rowBase, M);
}

// ---------------------------------------- GEMM1 (node): build fused A-tile
__global__ void node_gemm1(const float* __restrict__ node_rep,
                           const float* __restrict__ deg,
                           const float* __restrict__ S2,
                           const float* __restrict__ O,
                           const float* __restrict__ P,
                           const float* __restrict__ Q,
                           const float* __restrict__ W1,
                           float* __restrict__ Hn, long M) {
  extern __shared__ float tile[];  // 128 x ASTRIDE floats
  long rowBase = (long)blockIdx.x * 128;
  for (int idx = threadIdx.x; idx < 128 * 64; idx += 256) {
    int q = idx >> 6, d = idx & 63;
    long v = rowBase + q;
    float* Arow = &tile[(size_t)q * ASTRIDE];
    if (v < M) {
      float dg  = deg[v];
      float dg2 = dg * dg;
      float s2  = S2[(size_t)v * 64 + d];
      float o   = O[(size_t)v * 64 + d];
      float pp  = P[(size_t)v * 64 + d];
      float qq  = Q[(size_t)v * 64 + d];
      float nr  = node_rep[(size_t)v * 64 + d];
      float a0 = dg2 * s2;
      Arow[d]        = nr;                       // node_rep
      Arow[64 + d]   = a0;                       // e2n b0
      Arow[128 + d]  = a0 + dg * o;              // e2n b1
      Arow[192 + d]  = a0 + pp;                  // e2n b2
      Arow[256 + d]  = a0 + dg * o + pp + s2;    // e2n b3
      Arow[320 + d]  = dg2 * nr;                 // e2n b4
      Arow[384 + d]  = dg2 * nr + qq;            // e2n b5
    } else {
#pragma unroll
      for (int b = 0; b < 7; ++b) Arow[b * 64 + d] = 0.f;
    }
  }
  __syncthreads();
  wmma_tile_compute<8>(tile, ASTRIDE, 448, W1, Hn, rowBase, M);
}

// ------------- GEMM2: BN+ReLU fused into A-tile load of H, then H' @ W2 -> Z
__global__ void gemm_bn2(const float* __restrict__ Hbuf,
                         const float* __restrict__ scale,
                         const float* __restrict__ shift,
                         const float* __restrict__ W2,
                         float* __restrict__ Z, long M) {
  extern __shared__ float tile[];  // 128 x BSTRIDE floats
  long rowBase = (long)blockIdx.x * 128;
  for (int idx = threadIdx.x; idx < 128 * 128; idx += 256) {
    int q = idx >> 7, k = idx & 127;
    long r = rowBase + q;
    float h = (r < M) ? Hbuf[r * 128 + k] : 0.f;
    tile[(size_t)q * BSTRIDE + k] = fmaxf(0.f, fmaf(h, scale[k], shift[k]));
  }
  __syncthreads();
  wmma_tile_compute<4>(tile, BSTRIDE, 128, W2, Z, rowBase, M);
}

// --------------------------------------------------------- BN statistics
__global__ void colstats(const float* __restrict__ X, long M, int C,
                         int rowsPerBlock, float* __restrict__ sum,
                         float* __restrict__ sumsq) {
  __shared__ float sh1[256], sh2[256];
  int tid = threadIdx.x;
  int col = tid % C;
  int sub = tid / C;
  int lanesPerCol = 256 / C;
  long r0 = (long)blockIdx.x * rowsPerBlock + sub;
  long rend = (long)(blockIdx.x + 1) * rowsPerBlock;
  if (rend > M) rend = M;
  float s = 0.f, ss = 0.f;
  for (long r = r0; r < rend; r += lanesPerCol) {
    float x = X[r * C + col];
    s += x;
    ss += x * x;
  }
  sh1[tid] = s;
  sh2[tid] = ss;
  __syncthreads();
  if (tid < C) {
    for (int k = 1; k < lanesPerCol; ++k) {
      s += sh1[tid + k * C];
      ss += sh2[tid + k * C];
    }
    atomicAdd(&sum[col], s);
    atomicAdd(&sumsq[col], ss);
  }
}

__global__ void bnfinalize(const float* __restrict__ sum,
                           const float* __restrict__ sumsq,
                           const float* __restrict__ g,
                           const float* __restrict__ b, float invM, int C,
                           float* __restrict__ scale, float* __restrict__ shift) {
  int c = blockIdx.x * blockDim.x + threadIdx.x;
  if (c < C) {
    float m = sum[c] * invM;
    float v = fmaxf(sumsq[c] * invM - m * m, 0.f);
    float sc = g[c] * rsqrtf(v + 1e-5f);
    scale[c] = sc;
    shift[c] = b[c] - m * sc;
  }
}

__global__ void bnrelu_inplace(float* __restrict__ Z, long total, int C,
                               const float* __restrict__ scale,
                               const float* __restrict__ shift) {
  long i = (long)blockIdx.x * blockDim.x + threadIdx.x;
  if (i < total) {
    int c = (int)(i & (long)(C - 1));
    Z[i] = fmaxf(0.f, fmaf(Z[i], scale[c], shift[c]));
  }
}

// ------------------------------------------------------------------ driver
static inline unsigned cdivu(long a, long b) { return (unsigned)((a + b - 1) / b); }

extern "C" void kernel_launch(void* const* d_in, const int* in_sizes, int n_in,
                              void* d_out, int out_size, void* d_ws, size_t ws_size,
                              hipStream_t stream) {
  const float* node_rep = (const float*)d_in[0];
  const float* edge_rep = (const float*)d_in[1];
  const float* Wn1 = (const float*)d_in[2];
  const float* gn1 = (const float*)d_in[3];
  const float* bn1 = (const float*)d_in[4];
  const float* Wn2 = (const float*)d_in[5];
  const float* gn2 = (const float*)d_in[6];
  const float* bn2 = (const float*)d_in[7];
  const float* We1 = (const float*)d_in[8];
  const float* ge1 = (const float*)d_in[9];
  const float* be1 = (const float*)d_in[10];
  const float* We2 = (const float*)d_in[11];
  const float* ge2 = (const float*)d_in[12];
  const float* be2 = (const float*)d_in[13];
  const int* eidx  = (const int*)d_in[14];

  const long N  = in_sizes[0] / 64;   // 50000
  const long E2 = in_sizes[1] / 64;   // 800000
  const int  E  = (int)(E2 / 2);

  // workspace layout (floats)
  float* ws = (float*)d_ws;
  size_t off = 0;
  float* deg = ws + off;  off += (size_t)((N + 255) & ~255L);
  float* S2  = ws + off;  off += (size_t)N * 64;
  float* O   = ws + off;  off += (size_t)N * 64;
  float* P   = ws + off;  off += (size_t)N * 64;
  float* Q   = ws + off;  off += (size_t)N * 64;
  float* st  = ws + off;  off += 2048;           // 4 layers x (sum,sumsq,scale,shift)
  const size_t zeroCount = off;                  // everything above needs zeroing
  float* He  = ws + off;  off += (size_t)E2 * 128;
  float* Hn  = ws + off;  off += (size_t)N * 128;
  (void)ws_size; (void)n_in; (void)out_size;

  float *sumE1 = st,        *sqE1 = st + 128,  *scE1 = st + 256,  *shE1 = st + 384;
  float *sumN1 = st + 512,  *sqN1 = st + 640,  *scN1 = st + 768,  *shN1 = st + 896;
  float *sumE2 = st + 1024, *sqE2 = st + 1152, *scE2 = st + 1280, *shE2 = st + 1408;
  float *sumN2 = st + 1536, *sqN2 = st + 1664, *scN2 = st + 1792, *shN2 = st + 1920;

  float* node_out = (float*)d_out;                  // [N, 64]
  float* edge_out = node_out + (size_t)N * 64;      // [2E, 64]

  const size_t ldsA = (size_t)128 * ASTRIDE * sizeof(float);  // 231,424 B
  const size_t ldsB = (size_t)128 * BSTRIDE * sizeof(float);  // 67,584 B

  // 1) zero accumulators + stats
  zerof<<<cdivu((long)zeroCount, 256), 256, 0, stream>>>(ws, (long)zeroCount);
  // 2) scatter passes
  scatter1<<<cdivu(E2 * 64, 256), 256, 0, stream>>>(edge_rep, node_rep, eidx, E, E2,
                                                    deg, S2, O, Q);
  scatter2<<<cdivu(E2 * 64, 256), 256, 0, stream>>>(eidx, E, E2, S2, P);
  // 3) layer-1 GEMMs (fused gather -> LDS A-tile -> WMMA)
  edge_gemm1<<<cdivu(E2, 128), 256, ldsA, stream>>>(edge_rep, node_rep, eidx, E,
                                                    deg, S2, O, We1, He, E2);
  node_gemm1<<<cdivu(N, 128), 256, ldsA, stream>>>(node_rep, deg, S2, O, P, Q,
                                                   Wn1, Hn, N);
  // 4) BN1 stats
  colstats<<<cdivu(E2, 4096), 256, 0, stream>>>(He, E2, 128, 4096, sumE1, sqE1);
  colstats<<<cdivu(N, 4096), 256, 0, stream>>>(Hn, N, 128, 4096, sumN1, sqN1);
  bnfinalize<<<1, 128, 0, stream>>>(sumE1, sqE1, ge1, be1, 1.f / (float)E2, 128, scE1, shE1);
  bnfinalize<<<1, 128, 0, stream>>>(sumN1, sqN1, gn1, bn1, 1.f / (float)N, 128, scN1, shN1);
  // 5) layer-2 GEMMs (BN1+ReLU fused on A-load), raw Z straight into d_out
  gemm_bn2<<<cdivu(E2, 128), 256, ldsB, stream>>>(He, scE1, shE1, We2, edge_out, E2);
  gemm_bn2<<<cdivu(N, 128), 256, ldsB, stream>>>(Hn, scN1, shN1, Wn2, node_out, N);
  // 6) BN2 stats + in-place BN+ReLU
  colstats<<<cdivu(E2, 4096), 256, 0, stream>>>(edge_out, E2, 64, 4096, sumE2, sqE2);
  colstats<<<cdivu(N, 4096), 256, 0, stream>>>(node_out, N, 64, 4096, sumN2, sqN2);
  bnfinalize<<<1, 64, 0, stream>>>(sumE2, sqE2, ge2, be2, 1.f / (float)E2, 64, scE2, shE2);
  bnfinalize<<<1, 64, 0, stream>>>(sumN2, sqN2, gn2, bn2, 1.f / (float)N, 64, scN2, shN2);
  bnrelu_inplace<<<cdivu(E2 * 64, 256), 256, 0, stream>>>(edge_out, E2 * 64, 64, scE2, shE2);
  bnrelu_inplace<<<cdivu(N * 64, 256), 256, 0, stream>>>(node_out, N * 64, 64, scN2, shN2);
}